// Attention_14937896255839
// MI455X (gfx1250) — compile-verified
//
#include <hip/hip_runtime.h>
#include <hip/hip_bf16.h>
#include <math.h>

typedef float v2f __attribute__((ext_vector_type(2)));
typedef float v8f __attribute__((ext_vector_type(8)));

// D = A(16x4) * B(4x16) + C, all fp32, wave32.
#define WMMA4(acc, a, b)                                                       \
  acc = __builtin_amdgcn_wmma_f32_16x16x4_f32(false, (a), false, (b),          \
                                              (short)0, (acc), false, false)

static constexpr int Bb = 2;
static constexpr int Ss = 2048;
static constexpr int Hh = 16;
static constexpr int Ee = 64;
static constexpr int NE = 1024;       // N_EMBD
static constexpr int Mrows = Bb * Ss; // 4096

// ---------------------------------------------------------------------------
// Kernel 1: qkv = x @ W_qkv^T + b_qkv, scattered into Q/K/V as (B,H,S,E)
// One wave -> 16 rows x 64 cols. M=4096, N=3072, K=1024.
// ---------------------------------------------------------------------------
__global__ void __launch_bounds__(128) qkv_gemm_kernel(
    const float* __restrict__ x, const float* __restrict__ Wqkv,
    const float* __restrict__ bqkv, float* __restrict__ Q,
    float* __restrict__ K, float* __restrict__ V) {
  const int wid  = blockIdx.x * 4 + (threadIdx.x >> 5);
  const int lane = threadIdx.x & 31;
  const int half = lane >> 4;
  const int lr   = lane & 15;

  const int tile_m = wid / 48;   // 0..255
  const int tile_n = wid % 48;   // 0..47 (64 cols each)
  const int m0 = tile_m * 16;
  const int n0 = tile_n * 64;

  v8f acc0 = {}, acc1 = {}, acc2 = {}, acc3 = {};
  const float* arow = x    + (size_t)(m0 + lr) * NE + half * 2;
  const float* brow = Wqkv + (size_t)(n0 + lr) * NE + half * 2;

  for (int k0 = 0; k0 < NE; k0 += 4) {
    v2f a = *(const v2f*)(arow + k0);
    v2f b0 = *(const v2f*)(brow + k0);
    v2f b1 = *(const v2f*)(brow + 16 * NE + k0);
    v2f b2 = *(const v2f*)(brow + 32 * NE + k0);
    v2f b3 = *(const v2f*)(brow + 48 * NE + k0);
    WMMA4(acc0, a, b0);
    WMMA4(acc1, a, b1);
    WMMA4(acc2, a, b2);
    WMMA4(acc3, a, b3);
  }

  v8f accs[4] = {acc0, acc1, acc2, acc3};
#pragma unroll
  for (int j = 0; j < 4; ++j) {
    const int col  = n0 + j * 16 + lr;     // 0..3071
    const float bi = bqkv[col];
    const int t    = col >> 10;            // 0:Q 1:K 2:V
    const int rem  = col & 1023;
    const int h    = rem >> 6;
    const int e    = rem & 63;
    float* dst = (t == 0) ? Q : (t == 1) ? K : V;
#pragma unroll
    for (int v = 0; v < 8; ++v) {
      const int r = m0 + v + 8 * half;     // 0..4095
      const int b = r >> 11;
      const int s = r & 2047;
      dst[(((size_t)(b * Hh + h) * Ss + s) << 6) + e] = accs[j][v] + bi;
    }
  }
}

// ---------------------------------------------------------------------------
// Kernel 2: scores = (Q @ K^T) * 0.125 with causal mask -> attn buffer (raw)
// One wave -> 16 rows x 64 cols of one (b,h) score matrix. K dim = 64.
// ---------------------------------------------------------------------------
__global__ void __launch_bounds__(128) scores_kernel(
    const float* __restrict__ Q, const float* __restrict__ K,
    float* __restrict__ attn) {
  const int wid  = blockIdx.x * 4 + (threadIdx.x >> 5);
  const int lane = threadIdx.x & 31;
  const int half = lane >> 4;
  const int lr   = lane & 15;

  const int bh = wid / (128 * 32);        // 0..31
  const int tm = (wid / 32) % 128;        // 16-row tile
  const int tn = wid % 32;                // 64-col tile
  const int m0 = tm * 16;
  const int n0 = tn * 64;

  const float* Qbh = Q + (size_t)bh * Ss * Ee;
  const float* Kbh = K + (size_t)bh * Ss * Ee;
  float* Abh = attn + (size_t)bh * Ss * Ss;

  v8f acc0 = {}, acc1 = {}, acc2 = {}, acc3 = {};
  if (n0 <= m0) {  // tile touches or is below the diagonal -> compute
    const float* arow = Qbh + (size_t)(m0 + lr) * Ee + half * 2;
    const float* brow = Kbh + (size_t)(n0 + lr) * Ee + half * 2;
#pragma unroll
    for (int k0 = 0; k0 < Ee; k0 += 4) {
      v2f a  = *(const v2f*)(arow + k0);
      v2f b0 = *(const v2f*)(brow + k0);
      v2f b1 = *(const v2f*)(brow + 16 * Ee + k0);
      v2f b2 = *(const v2f*)(brow + 32 * Ee + k0);
      v2f b3 = *(const v2f*)(brow + 48 * Ee + k0);
      WMMA4(acc0, a, b0);
      WMMA4(acc1, a, b1);
      WMMA4(acc2, a, b2);
      WMMA4(acc3, a, b3);
    }
  }

  v8f accs[4] = {acc0, acc1, acc2, acc3};
#pragma unroll
  for (int j = 0; j < 4; ++j) {
    const int col = n0 + j * 16 + lr;
#pragma unroll
    for (int v = 0; v < 8; ++v) {
      const int row = m0 + v + 8 * half;
      const float val = (col > row) ? -10000.0f : accs[j][v] * 0.125f;
      Abh[(size_t)row * Ss + col] = val;
    }
  }
}

// ---------------------------------------------------------------------------
// Kernel 3: in-place row softmax over attn (rows of length 2048)
// One 256-thread block per row; each thread owns 8 elements.
// ---------------------------------------------------------------------------
__global__ void __launch_bounds__(256) softmax_kernel(float* __restrict__ attn) {
  __shared__ float red[256];
  const int tid = threadIdx.x;
  float* row = attn + (size_t)blockIdx.x * Ss;

  float v[8];
#pragma unroll
  for (int i = 0; i < 8; ++i) v[i] = row[tid + 256 * i];

  float mx = v[0];
#pragma unroll
  for (int i = 1; i < 8; ++i) mx = fmaxf(mx, v[i]);
  red[tid] = mx;
  __syncthreads();
  for (int off = 128; off > 0; off >>= 1) {
    if (tid < off) red[tid] = fmaxf(red[tid], red[tid + off]);
    __syncthreads();
  }
  mx = red[0];
  __syncthreads();

  float sum = 0.0f;
#pragma unroll
  for (int i = 0; i < 8; ++i) {
    v[i] = expf(v[i] - mx);
    sum += v[i];
  }
  red[tid] = sum;
  __syncthreads();
  for (int off = 128; off > 0; off >>= 1) {
    if (tid < off) red[tid] += red[tid + off];
    __syncthreads();
  }
  const float inv = 1.0f / red[0];

#pragma unroll
  for (int i = 0; i < 8; ++i) row[tid + 256 * i] = v[i] * inv;
}

// ---------------------------------------------------------------------------
// Kernel 4: z = attn @ V per (b,h); k-loop truncated at the causal diagonal.
// One wave -> 16 rows x 64 cols (all of E). Output in (B,S,H*E) layout.
// ---------------------------------------------------------------------------
__global__ void __launch_bounds__(128) pv_gemm_kernel(
    const float* __restrict__ attn, const float* __restrict__ V,
    float* __restrict__ Zr) {
  const int wid  = blockIdx.x * 4 + (threadIdx.x >> 5);
  const int lane = threadIdx.x & 31;
  const int half = lane >> 4;
  const int lr   = lane & 15;

  const int bh = wid / 128;     // 0..31
  const int tm = wid % 128;     // 16-row tile
  const int m0 = tm * 16;

  const float* Abh = attn + (size_t)bh * Ss * Ss;
  const float* Vbh = V + (size_t)bh * Ss * Ee;

  v8f acc0 = {}, acc1 = {}, acc2 = {}, acc3 = {};
  const float* arow = Abh + (size_t)(m0 + lr) * Ss + half * 2;
  const int kmax = m0 + 16;     // attn[row][k] == 0 for k > row
  for (int k0 = 0; k0 < kmax; k0 += 4) {
    v2f a = *(const v2f*)(arow + k0);
    const int kr0 = k0 + 2 * half;
    const float* vr0 = Vbh + (size_t)kr0 * Ee + lr;
    v2f b0, b1, b2, b3;
    b0.x = vr0[0];       b0.y = vr0[Ee];
    b1.x = vr0[16];      b1.y = vr0[Ee + 16];
    b2.x = vr0[32];      b2.y = vr0[Ee + 32];
    b3.x = vr0[48];      b3.y = vr0[Ee + 48];
    WMMA4(acc0, a, b0);
    WMMA4(acc1, a, b1);
    WMMA4(acc2, a, b2);
    WMMA4(acc3, a, b3);
  }

  const int b = bh / Hh;
  const int h = bh % Hh;
  v8f accs[4] = {acc0, acc1, acc2, acc3};
#pragma unroll
  for (int j = 0; j < 4; ++j) {
    const int e = j * 16 + lr;
#pragma unroll
    for (int v = 0; v < 8; ++v) {
      const int s = m0 + v + 8 * half;
      Zr[(size_t)(b * Ss + s) * NE + h * Ee + e] = accs[j][v];
    }
  }
}

// ---------------------------------------------------------------------------
// Kernel 5: out = Zr @ W_out^T + b_out.  M=4096, N=1024, K=1024.
// ---------------------------------------------------------------------------
__global__ void __launch_bounds__(128) out_proj_kernel(
    const float* __restrict__ Zr, const float* __restrict__ Wout,
    const float* __restrict__ bout, float* __restrict__ out) {
  const int wid  = blockIdx.x * 4 + (threadIdx.x >> 5);
  const int lane = threadIdx.x & 31;
  const int half = lane >> 4;
  const int lr   = lane & 15;

  const int tile_m = wid / 16;  // 0..255
  const int tile_n = wid % 16;  // 0..15 (64 cols each)
  const int m0 = tile_m * 16;
  const int n0 = tile_n * 64;

  v8f acc0 = {}, acc1 = {}, acc2 = {}, acc3 = {};
  const float* arow = Zr   + (size_t)(m0 + lr) * NE + half * 2;
  const float* brow = Wout + (size_t)(n0 + lr) * NE + half * 2;
  for (int k0 = 0; k0 < NE; k0 += 4) {
    v2f a  = *(const v2f*)(arow + k0);
    v2f b0 = *(const v2f*)(brow + k0);
    v2f b1 = *(const v2f*)(brow + 16 * NE + k0);
    v2f b2 = *(const v2f*)(brow + 32 * NE + k0);
    v2f b3 = *(const v2f*)(brow + 48 * NE + k0);
    WMMA4(acc0, a, b0);
    WMMA4(acc1, a, b1);
    WMMA4(acc2, a, b2);
    WMMA4(acc3, a, b3);
  }

  v8f accs[4] = {acc0, acc1, acc2, acc3};
#pragma unroll
  for (int j = 0; j < 4; ++j) {
    const int col  = n0 + j * 16 + lr;
    const float bi = bout[col];
#pragma unroll
    for (int v = 0; v < 8; ++v) {
      const int r = m0 + v + 8 * half;
      out[(size_t)r * NE + col] = accs[j][v] + bi;
    }
  }
}

// ---------------------------------------------------------------------------
extern "C" void kernel_launch(void* const* d_in, const int* in_sizes, int n_in,
                              void* d_out, int out_size, void* d_ws,
                              size_t ws_size, hipStream_t stream) {
  const float* x     = (const float*)d_in[0];
  const float* W_qkv = (const float*)d_in[1];
  const float* b_qkv = (const float*)d_in[2];
  const float* W_out = (const float*)d_in[3];
  const float* b_out = (const float*)d_in[4];

  float* out  = (float*)d_out;                       // (B,S,1024)
  float* attn = (float*)d_out + (size_t)Mrows * NE;  // (B,H,S,S)

  const size_t perBuf = (size_t)Bb * Hh * Ss * Ee;   // 4M floats = 16 MB
  float* Q  = (float*)d_ws;
  float* K  = Q + perBuf;
  float* V  = K + perBuf;
  float* Zr = V + perBuf;                            // (B,S,H*E) = 16 MB

  // 1) QKV projection + rearrange:  256*48 wave-tiles, 4 waves/block
  qkv_gemm_kernel<<<3072, 128, 0, stream>>>(x, W_qkv, b_qkv, Q, K, V);

  // 2) masked scores:  32 * 128 * 32 wave-tiles
  scores_kernel<<<32768, 128, 0, stream>>>(Q, K, attn);

  // 3) row softmax in place: one block per row
  softmax_kernel<<<(Bb * Hh * Ss), 256, 0, stream>>>(attn);

  // 4) z = attn @ V -> (B,S,H*E):  32 * 128 wave-tiles
  pv_gemm_kernel<<<1024, 128, 0, stream>>>(attn, V, Zr);

  // 5) output projection: 256*16 wave-tiles
  out_proj_kernel<<<1024, 128, 0, stream>>>(Zr, W_out, b_out, out);
}